// GATLayer_62869731279387
// MI455X (gfx1250) — compile-verified
//
#include <hip/hip_runtime.h>
#include <math.h>

#define D   64
#define H   8
#define DH  8
#define ED  16

typedef float v2f __attribute__((ext_vector_type(2)));
typedef float v8f __attribute__((ext_vector_type(8)));

// ---------------------------------------------------------------------------
// Kernel 1: fused node projections  q|k|v|x_r = x @ W{q,k,v,skip} + b
// f32 WMMA 16x16x4. grid = (ceil(N/16), 4 matrices), block = 128 (4 waves),
// wave w computes the 16x16 output tile at columns [16w, 16w+16).
// ---------------------------------------------------------------------------
__global__ __launch_bounds__(128)
void node_proj_wmma(const float* __restrict__ x,
                    const float* __restrict__ Wq, const float* __restrict__ bq,
                    const float* __restrict__ Wk, const float* __restrict__ bk,
                    const float* __restrict__ Wv, const float* __restrict__ bv,
                    const float* __restrict__ Ws, const float* __restrict__ bs,
                    float* __restrict__ q, float* __restrict__ k,
                    float* __restrict__ v, float* __restrict__ xr, int n)
{
    const float* Wlist[4] = {Wq, Wk, Wv, Ws};
    const float* blist[4] = {bq, bk, bv, bs};
    float*       olist[4] = {q, k, v, xr};
    const int which   = blockIdx.y;
    const float* W    = Wlist[which];
    const float* bias = blist[which];
    float*       out  = olist[which];

    const int wave  = threadIdx.x >> 5;
    const int lane  = threadIdx.x & 31;
    const int row0  = blockIdx.x * 16;
    const int col0  = wave * 16;
    const int l     = lane & 15;            // M (for A) / N (for B,C)
    const int khalf = (lane >> 4) << 1;     // lanes 16-31 carry K=2,3
    const int rbase = (lane >> 4) << 3;     // C rows r / r+8

    int arow = row0 + l; if (arow >= n) arow = n - 1;   // clamp tail reads
    const float* ap = x + (size_t)arow * D;

    v8f acc = {};
#pragma unroll
    for (int kk = 0; kk < D; kk += 4) {
        v2f a, b;
        a.x = ap[kk + khalf];
        a.y = ap[kk + khalf + 1];
        b.x = W[(kk + khalf)     * D + col0 + l];
        b.y = W[(kk + khalf + 1) * D + col0 + l];
        acc = __builtin_amdgcn_wmma_f32_16x16x4_f32(
                  false, a, false, b, (short)0, acc, false, false);
    }

    const float bcol = bias[col0 + l];
#pragma unroll
    for (int r = 0; r < 8; ++r) {
        int row = row0 + rbase + r;
        if (row < n) out[(size_t)row * D + col0 + l] = acc[r] + bcol;
    }
}

// order-preserving float<->uint encoding for atomic max
__device__ __forceinline__ unsigned f2ord(float f) {
    unsigned u = __float_as_uint(f);
    return (u & 0x80000000u) ? ~u : (u | 0x80000000u);
}
__device__ __forceinline__ float ord2f(unsigned u) {
    return (u & 0x80000000u) ? __uint_as_float(u ^ 0x80000000u)
                             : __uint_as_float(~u);
}

// ---------------------------------------------------------------------------
// Kernel 2: per-(edge,head) logits + segment max.  e-projection recomputed
// on the fly from LDS-cached We/be (avoids materializing E*64 floats).
// ---------------------------------------------------------------------------
__global__ __launch_bounds__(256)
void edge_logits(const int* __restrict__ ei, const float* __restrict__ attr,
                 const float* __restrict__ We, const float* __restrict__ be,
                 const float* __restrict__ q, const float* __restrict__ k,
                 float* __restrict__ logits, unsigned* __restrict__ maxbuf,
                 int nedges)
{
    __shared__ float sWe[ED * D];
    __shared__ float sbe[D];
    for (int i = threadIdx.x; i < ED * D; i += blockDim.x) sWe[i] = We[i];
    if (threadIdx.x < D) sbe[threadIdx.x] = be[threadIdx.x];
    __syncthreads();

    const int tid = blockIdx.x * blockDim.x + threadIdx.x;
    if (tid >= nedges * H) return;
    const int e = tid >> 3;
    const int h = tid & 7;
    const int src = ei[e];
    const int dst = ei[nedges + e];

    float a[ED];
#pragma unroll
    for (int j = 0; j < ED; ++j) a[j] = attr[(size_t)e * ED + j];

    const float* qd = q + (size_t)dst * D + h * DH;
    const float* ks = k + (size_t)src * D + h * DH;
    float logit = 0.f;
#pragma unroll
    for (int dh = 0; dh < DH; ++dh) {
        float ev = sbe[h * DH + dh];
#pragma unroll
        for (int j = 0; j < ED; ++j) ev += a[j] * sWe[j * D + h * DH + dh];
        logit += qd[dh] * (ks[dh] + ev);
    }
    logit *= 0.35355339059327373f;   // 1/sqrt(DH)
    logits[tid] = logit;
    atomicMax(&maxbuf[dst * H + h], f2ord(logit));
}

// ---------------------------------------------------------------------------
// Kernel 3: p = exp(logit - max[dst]) (in place) + atomic denom
// ---------------------------------------------------------------------------
__global__ __launch_bounds__(256)
void edge_softmax_p(const int* __restrict__ ei, float* __restrict__ logits,
                    const unsigned* __restrict__ maxbuf,
                    float* __restrict__ denom, int nedges)
{
    const int tid = blockIdx.x * blockDim.x + threadIdx.x;
    if (tid >= nedges * H) return;
    const int e = tid >> 3;
    const int h = tid & 7;
    const int dst = ei[nedges + e];
    const float m = ord2f(maxbuf[dst * H + h]);
    const float p = __expf(logits[tid] - m);
    logits[tid] = p;
    atomicAdd(&denom[dst * H + h], p);
}

// ---------------------------------------------------------------------------
// Kernel 4: out[dst] += alpha * (v[src] + e)   (e recomputed from LDS We)
// ---------------------------------------------------------------------------
__global__ __launch_bounds__(256)
void edge_aggregate(const int* __restrict__ ei, const float* __restrict__ attr,
                    const float* __restrict__ We, const float* __restrict__ be,
                    const float* __restrict__ v, const float* __restrict__ p,
                    const float* __restrict__ denom, float* __restrict__ outacc,
                    int nedges)
{
    __shared__ float sWe[ED * D];
    __shared__ float sbe[D];
    for (int i = threadIdx.x; i < ED * D; i += blockDim.x) sWe[i] = We[i];
    if (threadIdx.x < D) sbe[threadIdx.x] = be[threadIdx.x];
    __syncthreads();

    const int tid = blockIdx.x * blockDim.x + threadIdx.x;
    if (tid >= nedges * H) return;
    const int e = tid >> 3;
    const int h = tid & 7;
    const int src = ei[e];
    const int dst = ei[nedges + e];

    const float alpha = p[tid] / (denom[dst * H + h] + 1e-16f);

    float a[ED];
#pragma unroll
    for (int j = 0; j < ED; ++j) a[j] = attr[(size_t)e * ED + j];

    const float* vs = v + (size_t)src * D + h * DH;
    float* od = outacc + (size_t)dst * D + h * DH;
#pragma unroll
    for (int dh = 0; dh < DH; ++dh) {
        float ev = sbe[h * DH + dh];
#pragma unroll
        for (int j = 0; j < ED; ++j) ev += a[j] * sWe[j * D + h * DH + dh];
        atomicAdd(&od[dh], alpha * (vs[dh] + ev));
    }
}

// ---------------------------------------------------------------------------
// Kernel 5: beta-gated skip:  g = sigmoid([out, xr, out-xr] @ Wbeta),
// gated result written in place over outacc.
// ---------------------------------------------------------------------------
__global__ __launch_bounds__(256)
void node_gate(const float* __restrict__ xr, float* __restrict__ outacc,
               const float* __restrict__ Wbeta, int n)
{
    const int nidx = blockIdx.x * blockDim.x + threadIdx.x;
    if (nidx >= n) return;
    const float* o = outacc + (size_t)nidx * D;
    const float* r = xr + (size_t)nidx * D;
    float z = 0.f;
#pragma unroll
    for (int i = 0; i < D; ++i) {
        const float oi = o[i], ri = r[i];
        z += oi * Wbeta[i] + ri * Wbeta[D + i] + (oi - ri) * Wbeta[2 * D + i];
    }
    const float g = 1.f / (1.f + __expf(-z));
    float* w = outacc + (size_t)nidx * D;
#pragma unroll
    for (int i = 0; i < D; ++i)
        w[i] = g * r[i] + (1.f - g) * w[i];
}

// ---------------------------------------------------------------------------
// Kernel 6: per-column mean / rstd (64 blocks, tree reduction, no atomics)
// ---------------------------------------------------------------------------
__global__ __launch_bounds__(256)
void col_stats(const float* __restrict__ gated, float* __restrict__ stats, int n)
{
    const int c = blockIdx.x;
    float s = 0.f, ss = 0.f;
    for (int r = threadIdx.x; r < n; r += blockDim.x) {
        const float vv = gated[(size_t)r * D + c];
        s += vv; ss += vv * vv;
    }
    __shared__ float sh[256], sh2[256];
    sh[threadIdx.x] = s; sh2[threadIdx.x] = ss;
    __syncthreads();
    for (int off = 128; off > 0; off >>= 1) {
        if (threadIdx.x < off) {
            sh[threadIdx.x]  += sh[threadIdx.x + off];
            sh2[threadIdx.x] += sh2[threadIdx.x + off];
        }
        __syncthreads();
    }
    if (threadIdx.x == 0) {
        const float inv_n = 1.f / (float)n;
        const float mean  = sh[0] * inv_n;
        const float var   = sh2[0] * inv_n - mean * mean;
        stats[c]     = mean;
        stats[D + c] = rsqrtf(var + 1e-5f);
    }
}

// ---------------------------------------------------------------------------
// Kernel 7: BN affine + LeakyReLU(0.01) -> d_out
// ---------------------------------------------------------------------------
__global__ __launch_bounds__(256)
void bn_apply(const float* __restrict__ gated, const float* __restrict__ stats,
              const float* __restrict__ gamma, const float* __restrict__ beta,
              float* __restrict__ out, int n)
{
    const int idx = blockIdx.x * blockDim.x + threadIdx.x;
    if (idx >= n * D) return;
    const int c = idx & (D - 1);
    const float o = (gated[idx] - stats[c]) * stats[D + c] * gamma[c] + beta[c];
    out[idx] = o >= 0.f ? o : 0.01f * o;
}

// ---------------------------------------------------------------------------
extern "C" void kernel_launch(void* const* d_in, const int* in_sizes, int n_in,
                              void* d_out, int out_size, void* d_ws, size_t ws_size,
                              hipStream_t stream)
{
    const float* x     = (const float*)d_in[0];
    const int*   ei    = (const int*)  d_in[1];
    const float* attr  = (const float*)d_in[2];
    const float* Wq    = (const float*)d_in[3];
    const float* bq    = (const float*)d_in[4];
    const float* Wk    = (const float*)d_in[5];
    const float* bk    = (const float*)d_in[6];
    const float* Wv    = (const float*)d_in[7];
    const float* bv    = (const float*)d_in[8];
    const float* We    = (const float*)d_in[9];
    const float* be    = (const float*)d_in[10];
    const float* Ws    = (const float*)d_in[11];
    const float* bs    = (const float*)d_in[12];
    const float* Wbeta = (const float*)d_in[13];
    const float* gamma = (const float*)d_in[14];
    const float* bbn   = (const float*)d_in[15];
    float* out = (float*)d_out;

    const int n = in_sizes[0] / D;         // 50000
    const int e = in_sizes[2] / ED;        // 800000
    const size_t nd = (size_t)n * D;
    const size_t eh = (size_t)e * H;
    const size_t nh = (size_t)n * H;

    float* ws = (float*)d_ws;
    float*    q      = ws;                       // nd
    float*    k      = q + nd;                   // nd
    float*    v      = k + nd;                   // nd
    float*    xr     = v + nd;                   // nd
    float*    logits = xr + nd;                  // eh (becomes p in place)
    // ---- zeroed-every-launch region (atomics + stats) ----
    unsigned* maxbuf = (unsigned*)(logits + eh); // nh
    float*    denom  = (float*)(maxbuf + nh);    // nh
    float*    outacc = denom + nh;               // nd
    float*    stats  = outacc + nd;              // 2*D
    const size_t zero_bytes = (2 * nh + nd + 2 * D) * sizeof(float);
    hipMemsetAsync((void*)maxbuf, 0, zero_bytes, stream);

    // 1) node projections (WMMA f32)
    dim3 g1((n + 15) / 16, 4);
    node_proj_wmma<<<g1, 128, 0, stream>>>(x, Wq, bq, Wk, bk, Wv, bv, Ws, bs,
                                           q, k, v, xr, n);
    // 2) logits + segment max
    const int ethreads = (int)eh;
    const int eblocks  = (ethreads + 255) / 256;
    edge_logits<<<eblocks, 256, 0, stream>>>(ei, attr, We, be, q, k,
                                             logits, maxbuf, e);
    // 3) exp + denom
    edge_softmax_p<<<eblocks, 256, 0, stream>>>(ei, logits, maxbuf, denom, e);
    // 4) weighted aggregate
    edge_aggregate<<<eblocks, 256, 0, stream>>>(ei, attr, We, be, v, logits,
                                                denom, outacc, e);
    // 5) beta gate (in place over outacc)
    node_gate<<<(n + 255) / 256, 256, 0, stream>>>(xr, outacc, Wbeta, n);
    // 6) batchnorm stats
    col_stats<<<D, 256, 0, stream>>>(outacc, stats, n);
    // 7) BN affine + LeakyReLU
    bn_apply<<<((int)nd + 255) / 256, 256, 0, stream>>>(outacc, stats, gamma,
                                                        bbn, out, n);
}